// MaskVectorQuantize_19696720019633
// MI455X (gfx1250) — compile-verified
//
#include <hip/hip_runtime.h>
#include <hip/hip_bf16.h>

typedef float v2f __attribute__((ext_vector_type(2)));
typedef float v8f __attribute__((ext_vector_type(8)));

#define CDIM   256
#define HW     1024
#define NROW   16384     // B*H*W
#define KCB    16384     // codebook size
#define TILE_M 64        // rows per workgroup (4 WMMA row tiles)
#define RT     4         // row tiles per wave
#define LDA    260       // LDS row stride (floats): bank-conflict-free for A-frag reads
#define BETA   0.25f
#define XQ_ELEMS 4194304 // 16*256*1024

// ---------------- workspace layout (floats): [0]=S accum, [1]=mask accum, [16..16+K) = wnorm

__global__ void vq_zero_ws(float* ws) {
    if (threadIdx.x < 2) ws[threadIdx.x] = 0.0f;
}

// one wave per codeword: |w_k|^2
__global__ __launch_bounds__(128) void vq_wnorm(const float* __restrict__ w,
                                                float* __restrict__ wn) {
    int wave = (blockIdx.x * blockDim.x + threadIdx.x) >> 5;
    int lane = threadIdx.x & 31;
    if (wave >= KCB) return;
    const float* row = w + (size_t)wave * CDIM;
    float s = 0.0f;
    for (int c = lane * 2; c < CDIM; c += 64) {
        float2 v = *(const float2*)(row + c);
        s += v.x * v.x + v.y * v.y;
    }
    #pragma unroll
    for (int m = 16; m >= 1; m >>= 1) s += __shfl_xor(s, m, 32);
    if (lane == 0) wn[wave] = s;
}

// Main fused kernel: distance GEMM (fp32 WMMA, M64xN32 per wave-iter) + argmax + gather + loss
__global__ __launch_bounds__(128)
void MaskVectorQuantize_19696720019633_kernel(const float* __restrict__ x,
                                              const float* __restrict__ mask,
                                              const float* __restrict__ weight,
                                              const float* __restrict__ wnorm,
                                              float* __restrict__ out,
                                              float* __restrict__ acc) {
    extern __shared__ float As[];           // TILE_M x LDA staged x tile, xf[r][c]
    __shared__ float rv[4][TILE_M];         // per-wave best value per row
    __shared__ int   ri[4][TILE_M];         // per-wave best index per row
    __shared__ float redS[128];
    __shared__ float redM[128];

    const int tid  = threadIdx.x;
    const int w    = tid >> 5;
    const int lane = tid & 31;
    const int n    = lane & 15;             // WMMA column / A row within tile
    const int h    = lane >> 4;             // half-wave selector
    const int rowbase = blockIdx.x * TILE_M;
    const int b    = rowbase / HW;          // 64 | 1024, so whole tile in one batch
    const int pbase = rowbase % HW;

    // ---- stage x tile via CDNA5 async global->LDS DMA (ASYNCcnt-tracked, no VGPR round trip)
    // As[r][c] = x[b][c][pbase+r]; consecutive lanes -> consecutive p (contiguous global)
    {
        const unsigned asbase = (unsigned)(size_t)&As[0];
        for (int i = tid; i < TILE_M * CDIM; i += 128) {
            int r = i & (TILE_M - 1);
            int c = i >> 6;                                   // TILE_M == 64
            unsigned lds_addr = asbase + (unsigned)(r * LDA + c) * 4u;
            unsigned long long gaddr =
                (unsigned long long)(size_t)(x + ((size_t)b * CDIM + c) * HW + pbase + r);
            asm volatile("global_load_async_to_lds_b32 %0, %1, off"
                         :: "v"(lds_addr), "v"(gaddr) : "memory");
        }
        asm volatile("s_wait_asynccnt 0x0" ::: "memory");
    }
    __syncthreads();

    float bestv[RT][8];
    int   besti[RT][8];
    #pragma unroll
    for (int rt = 0; rt < RT; ++rt)
        #pragma unroll
        for (int r = 0; r < 8; ++r) { bestv[rt][r] = -3.4e38f; besti[rt][r] = 0; }

    // ---- GEMM + fused argmax. Each wave-iter: 2 adjacent column tiles (N=32) x 4 row tiles
    // (M=64); A fragments reused across both column tiles -> 128B LDS per wmma.
    const int NT = KCB / 16;                 // 1024 column tiles
    for (int ct0 = 2 * w; ct0 < NT; ct0 += 8) {
        const int colbase = ct0 << 4;        // tiles ct0 and ct0+1
        const float* __restrict__ wrow0 = weight + (size_t)(colbase + n) * CDIM;
        const float* __restrict__ wrow1 = wrow0 + (size_t)16 * CDIM;
        const float wna = wnorm[colbase + n];
        const float wnb = wnorm[colbase + 16 + n];
        // prefetch this lane's codeword rows for the next wave-iter (global_prefetch_b8)
        __builtin_prefetch(wrow0 + (size_t)128 * CDIM, 0, 1);
        __builtin_prefetch(wrow1 + (size_t)128 * CDIM, 0, 1);

        v8f c00 = {0.f,0.f,0.f,0.f,0.f,0.f,0.f,0.f};
        v8f c01 = c00, c02 = c00, c03 = c00;
        v8f c10 = c00, c11 = c00, c12 = c00, c13 = c00;
        #pragma unroll 2
        for (int c0 = 0; c0 < CDIM; c0 += 4) {
            v2f b0 = *(const v2f*)(wrow0 + c0 + 2 * h);                   // B frags (L2-resident)
            v2f b1 = *(const v2f*)(wrow1 + c0 + 2 * h);
            v2f a0 = *(const v2f*)(&As[(     n) * LDA + c0 + 2 * h]);     // rows  0..15
            v2f a1 = *(const v2f*)(&As[(16 + n) * LDA + c0 + 2 * h]);     // rows 16..31
            v2f a2 = *(const v2f*)(&As[(32 + n) * LDA + c0 + 2 * h]);     // rows 32..47
            v2f a3 = *(const v2f*)(&As[(48 + n) * LDA + c0 + 2 * h]);     // rows 48..63
            c00 = __builtin_amdgcn_wmma_f32_16x16x4_f32(false, a0, false, b0, (short)0, c00, false, false);
            c01 = __builtin_amdgcn_wmma_f32_16x16x4_f32(false, a1, false, b0, (short)0, c01, false, false);
            c02 = __builtin_amdgcn_wmma_f32_16x16x4_f32(false, a2, false, b0, (short)0, c02, false, false);
            c03 = __builtin_amdgcn_wmma_f32_16x16x4_f32(false, a3, false, b0, (short)0, c03, false, false);
            c10 = __builtin_amdgcn_wmma_f32_16x16x4_f32(false, a0, false, b1, (short)0, c10, false, false);
            c11 = __builtin_amdgcn_wmma_f32_16x16x4_f32(false, a1, false, b1, (short)0, c11, false, false);
            c12 = __builtin_amdgcn_wmma_f32_16x16x4_f32(false, a2, false, b1, (short)0, c12, false, false);
            c13 = __builtin_amdgcn_wmma_f32_16x16x4_f32(false, a3, false, b1, (short)0, c13, false, false);
        }
        const int colA = colbase + n;        // tile0 column (smaller index)
        const int colB = colA + 16;          // tile1 column
        #pragma unroll
        for (int r = 0; r < 8; ++r) {
            // tile0 first (lower index), strict > preserves first-max semantics
            float v;
            v = 2.0f * c00[r] - wna; if (v > bestv[0][r]) { bestv[0][r] = v; besti[0][r] = colA; }
            v = 2.0f * c01[r] - wna; if (v > bestv[1][r]) { bestv[1][r] = v; besti[1][r] = colA; }
            v = 2.0f * c02[r] - wna; if (v > bestv[2][r]) { bestv[2][r] = v; besti[2][r] = colA; }
            v = 2.0f * c03[r] - wna; if (v > bestv[3][r]) { bestv[3][r] = v; besti[3][r] = colA; }
            v = 2.0f * c10[r] - wnb; if (v > bestv[0][r]) { bestv[0][r] = v; besti[0][r] = colB; }
            v = 2.0f * c11[r] - wnb; if (v > bestv[1][r]) { bestv[1][r] = v; besti[1][r] = colB; }
            v = 2.0f * c12[r] - wnb; if (v > bestv[2][r]) { bestv[2][r] = v; besti[2][r] = colB; }
            v = 2.0f * c13[r] - wnb; if (v > bestv[3][r]) { bestv[3][r] = v; besti[3][r] = colB; }
        }
    }

    // ---- butterfly argmax across the 16 columns held across lanes (first-index tie-break)
    #pragma unroll
    for (int rt = 0; rt < RT; ++rt) {
        #pragma unroll
        for (int r = 0; r < 8; ++r) {
            float v = bestv[rt][r];
            int   idx = besti[rt][r];
            #pragma unroll
            for (int m = 1; m <= 8; m <<= 1) {
                float ov = __shfl_xor(v, m, 32);
                int   oi = __shfl_xor(idx, m, 32);
                if (ov > v || (ov == v && oi < idx)) { v = ov; idx = oi; }
            }
            if (n == 0) {
                int rr = rt * 16 + 8 * h + r;   // row within block tile
                rv[w][rr] = v;
                ri[w][rr] = idx;
            }
        }
    }
    __syncthreads();

    // ---- cross-wave reduce; write embedding indices
    if (tid < TILE_M) {
        float v = rv[0][tid];
        int idx  = ri[0][tid];
        #pragma unroll
        for (int ww = 1; ww < 4; ++ww) {
            float ov = rv[ww][tid];
            int   oi = ri[ww][tid];
            if (ov > v || (ov == v && oi < idx)) { v = ov; idx = oi; }
        }
        ri[0][tid] = idx;
        out[XQ_ELEMS + 1 + rowbase + tid] = (float)idx;   // embed_ind (b,h,w) flat
    }
    __syncthreads();

    // ---- gather x_q, transpose-store to (b,c,h,w), accumulate masked SSE
    const int r2   = tid & (TILE_M - 1);
    const int csub = tid >> 6;                 // 0..1
    const int p    = pbase + r2;
    const int bidx = ri[0][r2];
    const float mk = mask[b * HW + p];
    const float* __restrict__ wq = weight + (size_t)bidx * CDIM;
    float s = 0.0f;
    for (int c = csub; c < CDIM; c += 2) {
        float xq = wq[c];
        float xf = As[r2 * LDA + c];
        float d  = xq - xf;
        s += mk * d * d;
        out[((size_t)b * CDIM + c) * HW + p] = xq;  // straight-through == x_q numerically
    }
    redS[tid] = s;
    redM[tid] = (csub == 0) ? mk : 0.0f;
    __syncthreads();
    #pragma unroll
    for (int off = 64; off >= 1; off >>= 1) {
        if (tid < off) { redS[tid] += redS[tid + off]; redM[tid] += redM[tid + off]; }
        __syncthreads();
    }
    if (tid == 0) {
        atomicAdd(&acc[0], redS[0]);
        atomicAdd(&acc[1], redM[0]);
    }
}

__global__ void vq_loss(const float* __restrict__ acc, float* __restrict__ out) {
    if (threadIdx.x == 0)
        out[XQ_ELEMS] = (1.0f + BETA) * acc[0] / ((float)CDIM * acc[1]);
}

extern "C" void kernel_launch(void* const* d_in, const int* in_sizes, int n_in,
                              void* d_out, int out_size, void* d_ws, size_t ws_size,
                              hipStream_t stream) {
    const float* x      = (const float*)d_in[0];   // (16,256,32,32)
    const float* mask   = (const float*)d_in[1];   // (16,1,32,32)
    const float* weight = (const float*)d_in[2];   // (16384,256)
    float* out = (float*)d_out;                    // x_q (4194304) | loss (1) | embed_ind (16384)
    float* acc = (float*)d_ws;                     // [0]=S, [1]=sum_mask
    float* wn  = acc + 16;                         // K floats of codeword norms

    const size_t smem = (size_t)TILE_M * LDA * sizeof(float);   // 66,560 B dynamic LDS

    vq_zero_ws<<<1, 64, 0, stream>>>(acc);
    vq_wnorm<<<KCB / 4, 128, 0, stream>>>(weight, wn);
    MaskVectorQuantize_19696720019633_kernel<<<NROW / TILE_M, 128, smem, stream>>>(
        x, mask, weight, wn, out, acc);
    vq_loss<<<1, 32, 0, stream>>>(acc, out);
}